// MutualInformationLoss_33251636806046
// MI455X (gfx1250) — compile-verified
//
#include <hip/hip_runtime.h>
#include <hip/hip_bf16.h>
#include <math.h>

#define BINS   100
#define NBIN2  (BINS * BINS)
#define HDIM   128

typedef __attribute__((ext_vector_type(2))) float v2f;
typedef __attribute__((ext_vector_type(4))) float v4f;
typedef __attribute__((ext_vector_type(8))) float v8f;

// ---- monotone float<->uint key (order-preserving for all floats) ----
__device__ __forceinline__ unsigned f2key(float f) {
    unsigned b = __float_as_uint(f);
    return (b & 0x80000000u) ? ~b : (b | 0x80000000u);
}
__device__ __forceinline__ float key2f(unsigned k) {
    unsigned b = (k & 0x80000000u) ? (k & 0x7fffffffu) : ~k;
    return __uint_as_float(b);
}

// ---- init workspace: [0]=min key, [1]=max key, [4..4+10000) hist ----
__global__ void mi_init(unsigned* ws) {
    int i = blockIdx.x * blockDim.x + threadIdx.x;
    if (i == 0) { ws[0] = 0xFFFFFFFFu; ws[1] = 0u; }
    if (i < NBIN2) ws[4 + i] = 0u;
}

// ---- global min/max of distances (vectorized b128 loads) ----
__global__ __launch_bounds__(256) void mi_minmax(const float* __restrict__ d,
                                                 int n, unsigned* ws) {
    __shared__ unsigned lmin, lmax;
    if (threadIdx.x == 0) { lmin = 0xFFFFFFFFu; lmax = 0u; }
    __syncthreads();
    unsigned kmin = 0xFFFFFFFFu, kmax = 0u;
    const int n4 = n >> 2;
    const v4f* d4 = (const v4f*)d;
    for (int i = blockIdx.x * blockDim.x + threadIdx.x; i < n4;
         i += gridDim.x * blockDim.x) {
        v4f v = d4[i];
        #pragma unroll
        for (int u = 0; u < 4; ++u) {
            unsigned k = f2key(v[u]);
            kmin = min(kmin, k);
            kmax = max(kmax, k);
        }
    }
    // scalar tail
    for (int i = (n4 << 2) + blockIdx.x * blockDim.x + threadIdx.x; i < n;
         i += gridDim.x * blockDim.x) {
        unsigned k = f2key(d[i]);
        kmin = min(kmin, k);
        kmax = max(kmax, k);
    }
    atomicMin(&lmin, kmin);
    atomicMax(&lmax, kmax);
    __syncthreads();
    if (threadIdx.x == 0) { atomicMin(&ws[0], lmin); atomicMax(&ws[1], lmax); }
}

// searchsorted(edges, x, side='right') - 1 with edges[i] = lo + i*step
__device__ __forceinline__ int edge_bin(float x, float lo, float step, float inv_step) {
    int i = (int)floorf((x - lo) * inv_step);
    i = max(-1, min(BINS, i));
    while (i < BINS && x >= lo + (float)(i + 1) * step) ++i;
    while (i >= 0 && x < lo + (float)i * step) --i;
    return i;
}

// pick c[m] for m in 0..7 without dynamic VGPR indexing
__device__ __forceinline__ float sel8(v8f c, int m) {
    float s = c[0];
    s = (m == 1) ? c[1] : s;
    s = (m == 2) ? c[2] : s;
    s = (m == 3) ? c[3] : s;
    s = (m == 4) ? c[4] : s;
    s = (m == 5) ? c[5] : s;
    s = (m == 6) ? c[6] : s;
    s = (m == 7) ? c[7] : s;
    return s;
}

// ---- main streaming pass: WMMA row-reduction + LDS-privatized 2D histogram ----
// Software-pipelined: batches of 8 K-chunks (16 b64 loads) double-buffered
// against the 24 WMMAs of the previous batch; compiler modulo-schedules this
// into a rolling ~8-deep load pipeline per wave. Grid sized (384x16 waves)
// so aggregate demand comfortably exceeds the 23.3 TB/s HBM roofline.
__global__ __launch_bounds__(512) void mi_hist(const float* __restrict__ e1,
                                               const float* __restrict__ e2,
                                               const float* __restrict__ dist,
                                               int nrows, unsigned* ws) {
    __shared__ unsigned lhist[NBIN2];
    for (int i = threadIdx.x; i < NBIN2; i += blockDim.x) lhist[i] = 0u;
    __syncthreads();

    const float dmin  = key2f(ws[0]);
    const float dmax  = key2f(ws[1]);
    const float dstep = (dmax - dmin) * (1.0f / BINS);
    const float dinv  = (dstep > 0.0f) ? 1.0f / dstep : 0.0f;
    const float sstep = 2.0f / BINS;
    const float sinv  = (float)BINS * 0.5f;

    unsigned* ghist = ws + 4;

    const int lane = threadIdx.x & 31;
    const int wib  = threadIdx.x >> 5;
    const int wpb  = blockDim.x >> 5;
    const int gw   = blockIdx.x * wpb + wib;
    const int nw   = gridDim.x * wpb;
    const int ntiles = nrows >> 4;  // 16 rows per wave tile

    // A-matrix (16x4 f32) layout: lanes 0-15 hold K=0,1; lanes 16-31 hold K=2,3
    const int  mrow = lane & 15;
    const long koff = (long)((lane >> 4) << 1);  // 0 or 2
    const v2f  ones = {1.0f, 1.0f};              // B = 4x16 ones (layout-free)
    const long tile_stride = (long)nw << (4 + 7);  // floats between a wave's tiles

    for (int t = gw; t < ntiles; t += nw) {
        const long row = ((long)t << 4) + mrow;
        const float* p1 = e1 + (row << 7) + koff;
        const float* p2 = e2 + (row << 7) + koff;

        // prefetch this wave's next tile (speculative; dropped if OOB)
        __builtin_prefetch(p1 + tile_stride, 0, 1);
        __builtin_prefetch(p2 + tile_stride, 0, 1);

        v8f cdot = {}; v8f cn1 = {}; v8f cn2 = {};

        v2f xa[8], ya[8], xb[8], yb[8];
        #pragma unroll
        for (int u = 0; u < 8; ++u) {               // prologue: batch 0 loads
            xa[u] = *(const v2f*)(p1 + ((long)u << 2));
            ya[u] = *(const v2f*)(p2 + ((long)u << 2));
        }
        #pragma unroll
        for (int jb = 0; jb < (HDIM / 4); jb += 8) {
            if (jb + 8 < (HDIM / 4)) {              // issue next batch's loads
                #pragma unroll
                for (int u = 0; u < 8; ++u) {
                    xb[u] = *(const v2f*)(p1 + ((long)(jb + 8 + u) << 2));
                    yb[u] = *(const v2f*)(p2 + ((long)(jb + 8 + u) << 2));
                }
            }
            #pragma unroll
            for (int u = 0; u < 8; ++u) {           // consume current batch
                // D[m][n] += sum_k A[m][k]*1 -> exact f32 row sums, bcast over N
                cdot = __builtin_amdgcn_wmma_f32_16x16x4_f32(
                    false, xa[u] * ya[u], false, ones, (short)0, cdot, false, false);
                cn1 = __builtin_amdgcn_wmma_f32_16x16x4_f32(
                    false, xa[u] * xa[u], false, ones, (short)0, cn1, false, false);
                cn2 = __builtin_amdgcn_wmma_f32_16x16x4_f32(
                    false, ya[u] * ya[u], false, ones, (short)0, cn2, false, false);
            }
            #pragma unroll
            for (int u = 0; u < 8; ++u) { xa[u] = xb[u]; ya[u] = yb[u]; }
        }

        // C/D layout: VGPR v holds M=v (lanes 0-15) and M=v+8 (lanes 16-31).
        int src = (lane < 8) ? lane : ((lane < 16) ? (lane + 16) : lane);
        float sdot = __shfl(sel8(cdot, lane & 7), src, 32);
        float sn1  = __shfl(sel8(cn1,  lane & 7), src, 32);
        float sn2  = __shfl(sel8(cn2,  lane & 7), src, 32);

        if (lane < 16) {
            float r1  = fmaxf(sqrtf(sn1), 1e-12f);
            float r2  = fmaxf(sqrtf(sn2), 1e-12f);
            float sim = sdot / (r1 * r2);
            int is = edge_bin(sim, -1.0f, sstep, sinv);
            float dv = dist[row];
            int id = edge_bin(dv, dmin, dstep, dinv);
            if (is >= 0 && is < BINS && id >= 0 && id < BINS)
                atomicAdd(&lhist[is * BINS + id], 1u);
        }
    }
    __syncthreads();
    for (int i = threadIdx.x; i < NBIN2; i += blockDim.x) {
        unsigned c = lhist[i];
        if (c) atomicAdd(&ghist[i], c);
    }
}

// ---- finalize: counts -> normalized mutual information (single block) ----
__global__ __launch_bounds__(256) void mi_final(const unsigned* __restrict__ ws,
                                                float* __restrict__ out) {
    const unsigned* hist = ws + 4;
    __shared__ float red[256];
    __shared__ float rowm[BINS], colm[BINS];
    __shared__ float scal[3];
    const int tid = threadIdx.x;

    float t = 0.f;
    for (int i = tid; i < NBIN2; i += 256) t += (float)hist[i];
    red[tid] = t; __syncthreads();
    for (int s = 128; s > 0; s >>= 1) { if (tid < s) red[tid] += red[tid + s]; __syncthreads(); }
    if (tid == 0) scal[0] = red[0];
    __syncthreads();
    const float inv = 1.0f / scal[0];

    if (tid < BINS) {
        float rs = 0.f, cs = 0.f;
        for (int j = 0; j < BINS; ++j) {
            rs += (float)hist[tid * BINS + j];
            cs += (float)hist[j * BINS + tid];
        }
        rowm[tid] = rs * inv;
        colm[tid] = cs * inv;
    }
    __syncthreads();

    const float eps = 1e-10f;
    float mi = 0.f;
    for (int k = tid; k < NBIN2; k += 256) {
        float p = (float)hist[k] * inv;
        int i = k / BINS, j = k - i * BINS;
        mi += p * (logf(p + eps) - logf(rowm[i] + eps) - logf(colm[j] + eps));
    }
    red[tid] = mi; __syncthreads();
    for (int s = 128; s > 0; s >>= 1) { if (tid < s) red[tid] += red[tid + s]; __syncthreads(); }
    if (tid == 0) scal[1] = red[0];
    __syncthreads();

    red[tid] = (tid < BINS) ? -logf(rowm[tid] + eps) : 0.f;
    __syncthreads();
    for (int s = 128; s > 0; s >>= 1) { if (tid < s) red[tid] += red[tid + s]; __syncthreads(); }
    if (tid == 0) scal[2] = red[0];
    __syncthreads();

    red[tid] = (tid < BINS) ? -logf(colm[tid] + eps) : 0.f;
    __syncthreads();
    for (int s = 128; s > 0; s >>= 1) { if (tid < s) red[tid] += red[tid + s]; __syncthreads(); }
    if (tid == 0) out[0] = scal[1] / fminf(scal[2], red[0]);
}

extern "C" void kernel_launch(void* const* d_in, const int* in_sizes, int n_in,
                              void* d_out, int out_size, void* d_ws, size_t ws_size,
                              hipStream_t stream) {
    const float* emb1 = (const float*)d_in[0];
    const float* emb2 = (const float*)d_in[1];
    const float* dist = (const float*)d_in[2];
    float* out  = (float*)d_out;
    unsigned* ws = (unsigned*)d_ws;
    const int n = in_sizes[2];  // number of rows B

    mi_init<<<(NBIN2 + 255) / 256, 256, 0, stream>>>(ws);
    mi_minmax<<<256, 256, 0, stream>>>(dist, n, ws);
    mi_hist<<<384, 512, 0, stream>>>(emb1, emb2, dist, n, ws);
    mi_final<<<1, 256, 0, stream>>>(ws, out);
}